// GATAggregateGraphPooler_19146964205615
// MI455X (gfx1250) — compile-verified
//
#include <hip/hip_runtime.h>
#include <math.h>

#define NN 51200
#define EE 409600
#define NG 64
#define NPG 800

typedef float v2f __attribute__((ext_vector_type(2)));
typedef float v8f __attribute__((ext_vector_type(8)));

__device__ __forceinline__ unsigned fenc(float x){
  unsigned u = __float_as_uint(x);
  return (u & 0x80000000u) ? ~u : (u | 0x80000000u);
}
__device__ __forceinline__ float fdec(unsigned u){
  unsigned b = (u & 0x80000000u) ? (u & 0x7FFFFFFFu) : ~u;
  return __uint_as_float(b);
}
__device__ __forceinline__ float lrelu(float x){ return x >= 0.f ? x : 0.2f*x; }

// ---------------------------------------------------------------------------
// WMMA f32 GEMM: C[N x M] = A[N x K] @ B[K x M] (+bias). One 16x16 tile/wave.
// grid = (N/16, M/64), block = 128 (4 waves). A staged in padded LDS.
// ---------------------------------------------------------------------------
__global__ void wmma_gemm_f32(const float* __restrict__ A, const float* __restrict__ B,
                              const float* __restrict__ bias, float* __restrict__ C,
                              int K, int M, int lda, int ldc)
{
  __shared__ float sA[16][129];
  const int rowTile = blockIdx.x * 16;
  const int colBase = blockIdx.y * 64;
  const int t = threadIdx.x;
  const int wave = t >> 5;
  const int lane = t & 31;
  for (int i = t; i < 16 * K; i += 128) {
    int r = i / K, c = i - r * K;
    sA[r][c] = A[(size_t)(rowTile + r) * lda + c];
  }
  __syncthreads();
  const int m    = lane & 15;
  const int half = lane >> 4;
  const int col  = colBase + wave * 16 + m;
  v8f acc = {0.f,0.f,0.f,0.f,0.f,0.f,0.f,0.f};
  for (int k = 0; k < K; k += 4) {
    const int ka = k + 2 * half;
    v2f a; a.x = sA[m][ka];            a.y = sA[m][ka + 1];
    v2f b; b.x = B[(size_t)ka * M + col]; b.y = B[(size_t)(ka + 1) * M + col];
    acc = __builtin_amdgcn_wmma_f32_16x16x4_f32(false, a, false, b,
                                                (short)0, acc, false, false);
  }
  const float bv = bias ? bias[col] : 0.0f;
#pragma unroll
  for (int v = 0; v < 8; ++v) {
    int r = rowTile + v + 8 * half;
    C[(size_t)r * ldc + col] = acc[v] + bv;
  }
}

// ---- attention source/dest dots: asrc[n,h]=<h[n,h,:],att_src[h,:]> ---------
__global__ void att_dots(const float* __restrict__ H, const float* __restrict__ as_,
                         const float* __restrict__ ad_, float* __restrict__ asrc,
                         float* __restrict__ adst)
{
  int i = blockIdx.x * blockDim.x + threadIdx.x;
  if (i >= NN * 4) return;
  int n = i >> 2, h = i & 3;
  const float* hp = H + (size_t)n * 256 + h * 64;
  const float* s  = as_ + h * 64;
  const float* d  = ad_ + h * 64;
  float vs = 0.f, vd = 0.f;
  for (int c = 0; c < 64; ++c) { float v = hp[c]; vs += v * s[c]; vd += v * d[c]; }
  asrc[i] = vs; adst[i] = vd;
}

// ---- weff[d,h] = sum_c We[d, h*64+c] * att_e[h,c]  (64x4) ------------------
__global__ void weff_kernel(const float* __restrict__ We, const float* __restrict__ ae,
                            float* __restrict__ weff)
{
  int t = threadIdx.x;  // 0..255
  int d = t >> 2, h = t & 3;
  float s = 0.f;
  for (int c = 0; c < 64; ++c) s += We[(size_t)d * 256 + h * 64 + c] * ae[h * 64 + c];
  weff[d * 4 + h] = s;
}

// ---- per-edge: alpha = leaky(asrc[src]+adst[dst]+eattr.weff); atomic max ---
__global__ void edge_alpha_max(const float* __restrict__ eattr, const float* __restrict__ weff,
                               const int* __restrict__ src, const int* __restrict__ dst,
                               const float* __restrict__ asrc, const float* __restrict__ adst,
                               float* __restrict__ alpha, unsigned* __restrict__ mx)
{
  __shared__ float sw[256];
  sw[threadIdx.x] = weff[threadIdx.x];
  __syncthreads();
  int e = blockIdx.x * blockDim.x + threadIdx.x;
  if (e >= EE) return;
  int s = src[e], d = dst[e];
  float ea[4] = {0.f, 0.f, 0.f, 0.f};
  const float* row = eattr + (size_t)e * 64;
  for (int k = 0; k < 64; ++k) {
    float v = row[k];
    ea[0] += v * sw[k*4+0]; ea[1] += v * sw[k*4+1];
    ea[2] += v * sw[k*4+2]; ea[3] += v * sw[k*4+3];
  }
#pragma unroll
  for (int h = 0; h < 4; ++h) {
    float a = lrelu(asrc[s*4+h] + adst[d*4+h] + ea[h]);
    alpha[(size_t)e*4+h] = a;
    atomicMax(&mx[d*4+h], fenc(a));
  }
}

__global__ void edge_expsum(float* __restrict__ alpha, const int* __restrict__ dst,
                            const unsigned* __restrict__ mx, float* __restrict__ sum)
{
  int t = blockIdx.x * blockDim.x + threadIdx.x;
  if (t >= EE * 4) return;
  int e = t >> 2, h = t & 3;
  int d = dst[e];
  float w = expf(alpha[t] - fdec(mx[d*4+h]));
  alpha[t] = w;
  atomicAdd(&sum[d*4+h], w);
}

__global__ void edge_aggregate(const float* __restrict__ w, const float* __restrict__ sum,
                               const int* __restrict__ src, const int* __restrict__ dst,
                               const float* __restrict__ H, float* __restrict__ agg)
{
  size_t t = (size_t)blockIdx.x * blockDim.x + threadIdx.x;
  if (t >= (size_t)EE * 64) return;
  int e = (int)(t >> 6);
  int q = (int)(t & 63);          // float4 index into the 256-wide row
  int h = q >> 4;
  int s = src[e], d = dst[e];
  float ww = w[(size_t)e*4+h] / (sum[d*4+h] + 1e-16f);
  const float4 hv = *(const float4*)(H + (size_t)s * 256 + q * 4);
  float* ap = agg + (size_t)d * 256 + q * 4;
  atomicAdd(ap + 0, ww * hv.x);
  atomicAdd(ap + 1, ww * hv.y);
  atomicAdd(ap + 2, ww * hv.z);
  atomicAdd(ap + 3, ww * hv.w);
}

__global__ void x1_kernel(const float* __restrict__ agg, const float* __restrict__ bias,
                          float* __restrict__ x1)
{
  int t = blockIdx.x * blockDim.x + threadIdx.x;
  if (t >= NN * 64) return;
  int n = t >> 6, c = t & 63;
  const float* r = agg + (size_t)n * 256 + c;
  x1[t] = 0.25f * (r[0] + r[64] + r[128] + r[192]) + bias[c];
}

__global__ void x2_kernel(const float* __restrict__ agg, const float* __restrict__ bias,
                          float* __restrict__ x2)
{
  size_t t = (size_t)blockIdx.x * blockDim.x + threadIdx.x;
  if (t >= (size_t)NN * 256) return;
  x2[t] = agg[t] + bias[t & 255];
}

// --------------------------- stage C: pooling GAT ---------------------------
__global__ void hp_kernel(const float* __restrict__ x2, const float* __restrict__ Wp,
                          int hd, float* __restrict__ hp)
{
  __shared__ float sw[64];
  if (threadIdx.x < 64) sw[threadIdx.x] = Wp[threadIdx.x];
  __syncthreads();
  int n = blockIdx.x * blockDim.x + threadIdx.x;
  if (n >= NN) return;
  const float* r = x2 + (size_t)n * 256 + hd * 64;
  float s = 0.f;
  for (int c = 0; c < 64; ++c) s += r[c] * sw[c];
  hp[n] = s;
}

__global__ void pool_edge_max(const int* __restrict__ ei0, const int* __restrict__ ei1,
                              const float* __restrict__ hp, const float* __restrict__ asp,
                              const float* __restrict__ adp, float* __restrict__ we,
                              unsigned* __restrict__ mx)
{
  int e = blockIdx.x * blockDim.x + threadIdx.x;
  if (e >= EE) return;
  int s = ei0[e], d = ei1[e];
  float a = lrelu(asp[0] * hp[s] + adp[0] * hp[d]);
  we[e] = a;
  atomicMax(&mx[d], fenc(a));
}

__global__ void pool_self_max(const float* __restrict__ hp, const float* __restrict__ asp,
                              const float* __restrict__ adp, float* __restrict__ ws,
                              unsigned* __restrict__ mx)
{
  int n = blockIdx.x * blockDim.x + threadIdx.x;
  if (n >= NN) return;
  float a = lrelu((asp[0] + adp[0]) * hp[n]);
  ws[n] = a;
  atomicMax(&mx[n], fenc(a));
}

__global__ void pool_edge_expsum(float* __restrict__ we, const int* __restrict__ ei1,
                                 const unsigned* __restrict__ mx, float* __restrict__ sum)
{
  int e = blockIdx.x * blockDim.x + threadIdx.x;
  if (e >= EE) return;
  int d = ei1[e];
  float w = expf(we[e] - fdec(mx[d]));
  we[e] = w;
  atomicAdd(&sum[d], w);
}

__global__ void pool_self_expsum(float* __restrict__ ws, const unsigned* __restrict__ mx,
                                 float* __restrict__ sum)
{
  int n = blockIdx.x * blockDim.x + threadIdx.x;
  if (n >= NN) return;
  float w = expf(ws[n] - fdec(mx[n]));
  ws[n] = w;
  atomicAdd(&sum[n], w);
}

__global__ void pool_score_init(const float* __restrict__ hp, const float* __restrict__ ws,
                                const float* __restrict__ sum, const float* __restrict__ biasp,
                                float* __restrict__ score)
{
  int n = blockIdx.x * blockDim.x + threadIdx.x;
  if (n >= NN) return;
  score[n] = biasp[0] + ws[n] / (sum[n] + 1e-16f) * hp[n];
}

__global__ void pool_score_edges(const float* __restrict__ we, const float* __restrict__ sum,
                                 const int* __restrict__ ei0, const int* __restrict__ ei1,
                                 const float* __restrict__ hp, float* __restrict__ score)
{
  int e = blockIdx.x * blockDim.x + threadIdx.x;
  if (e >= EE) return;
  int s = ei0[e], d = ei1[e];
  atomicAdd(&score[d], we[e] / (sum[d] + 1e-16f) * hp[s]);
}

__global__ void edge_score_out(const float* __restrict__ score, const int* __restrict__ ei0,
                               const int* __restrict__ ei1, float* __restrict__ out, int hd)
{
  int e = blockIdx.x * blockDim.x + threadIdx.x;
  if (e >= EE) return;
  out[(size_t)e * 4 + hd] = 0.5f * (score[ei0[e]] + score[ei1[e]]);
}

// Bitonic per-graph stable-equivalent sort: descending score, idx tiebreak.
__global__ void seg_sort(const float* __restrict__ score, int* __restrict__ perm,
                         int* __restrict__ inv)
{
  __shared__ float sk[1024];
  __shared__ int   si[1024];
  int g = blockIdx.x, t = threadIdx.x;
  sk[t] = (t < NPG) ? score[g * NPG + t] : -INFINITY;
  si[t] = t;
  __syncthreads();
  for (int k = 2; k <= 1024; k <<= 1) {
    for (int j = k >> 1; j > 0; j >>= 1) {
      int x = t ^ j;
      if (x > t) {
        bool up = ((t & k) == 0);
        float s1 = sk[t], s2 = sk[x];
        int   i1 = si[t], i2 = si[x];
        bool aFirst = (s1 > s2) || (s1 == s2 && i1 < i2);
        if (up ? !aFirst : aFirst) { sk[t] = s2; sk[x] = s1; si[t] = i2; si[x] = i1; }
      }
      __syncthreads();
    }
  }
  if (t < NPG) {
    int p = si[t];
    perm[g * NPG + t] = g * NPG + p;   // global index of rank-t node
    inv[g * NPG + p]  = g * NPG + t;   // global new index of current node p
  }
}

__global__ void xhead_kernel(const float* __restrict__ x2, const float* __restrict__ score,
                             const int* __restrict__ perm, int hd, float* __restrict__ xh)
{
  size_t t = (size_t)blockIdx.x * blockDim.x + threadIdx.x;
  if (t >= (size_t)NN * 64) return;
  int j = (int)(t >> 6), c = (int)(t & 63);
  int p = perm[j];
  xh[t] = x2[(size_t)p * 256 + hd * 64 + c] * score[p];
}

__global__ void ei_update(int* __restrict__ ei, const int* __restrict__ inv)
{
  int t = blockIdx.x * blockDim.x + threadIdx.x;
  if (t >= 2 * EE) return;
  ei[t] = inv[ei[t]];
}

// -------- batchnorm stats (deterministic two-stage) + apply -----------------
__global__ void col_stats_partial(const float* __restrict__ Y, float* __restrict__ part)
{
  __shared__ float s1[256], s2[256];
  int col = threadIdx.x & 63, rr = threadIdx.x >> 6;
  int r0 = blockIdx.x * 256;
  float a = 0.f, b = 0.f;
  for (int r = r0 + rr; r < r0 + 256; r += 4) {
    float v = Y[(size_t)r * 64 + col];
    a += v; b += v * v;
  }
  s1[threadIdx.x] = a; s2[threadIdx.x] = b;
  __syncthreads();
  if (rr == 0) {
    for (int i = 1; i < 4; ++i) { a += s1[col + i * 64]; b += s2[col + i * 64]; }
    part[blockIdx.x * 128 + col]      = a;
    part[blockIdx.x * 128 + 64 + col] = b;
  }
}

__global__ void bn_finalize(const float* __restrict__ part, float* __restrict__ bn)
{
  int c = threadIdx.x;  // 64 threads
  float s = 0.f, q = 0.f;
  for (int b = 0; b < 200; ++b) { s += part[b * 128 + c]; q += part[b * 128 + 64 + c]; }
  float mean = s / (float)NN;
  float var  = q / (float)NN - mean * mean;
  bn[c]      = mean;
  bn[64 + c] = rsqrtf(var + 1e-5f);
}

__global__ void bn_apply(const float* __restrict__ Y, const float* __restrict__ bn,
                         const float* __restrict__ gamma, const float* __restrict__ beta,
                         float* __restrict__ out)
{
  size_t t = (size_t)blockIdx.x * blockDim.x + threadIdx.x;
  if (t >= (size_t)NN * 64) return;
  int c = (int)(t & 63);
  float v = (Y[t] - bn[c]) * bn[64 + c] * gamma[c] + beta[c];
  out[t] = v > 0.f ? v : 0.f;
}

// -------- per-graph softmax pooling:  out[g,hd,:] = sum alpha*emb ----------
__global__ void graph_pool(const float* __restrict__ gate, const float* __restrict__ emb,
                           float* __restrict__ out, int hd)
{
  __shared__ float red[8][64];
  int g = blockIdx.x, c = threadIdx.x, rr = threadIdx.y;
  size_t base = (size_t)g * NPG * 64;
  float m = -INFINITY;
  for (int r = rr; r < NPG; r += 8) m = fmaxf(m, gate[base + (size_t)r * 64 + c]);
  red[rr][c] = m; __syncthreads();
  if (rr == 0) { for (int i = 1; i < 8; ++i) m = fmaxf(m, red[i][c]); red[0][c] = m; }
  __syncthreads();
  m = red[0][c];
  __syncthreads();
  float s = 0.f;
  for (int r = rr; r < NPG; r += 8) s += expf(gate[base + (size_t)r * 64 + c] - m);
  red[rr][c] = s; __syncthreads();
  if (rr == 0) { for (int i = 1; i < 8; ++i) s += red[i][c]; red[0][c] = s; }
  __syncthreads();
  s = red[0][c];
  __syncthreads();
  float acc = 0.f;
  for (int r = rr; r < NPG; r += 8) {
    size_t o = base + (size_t)r * 64 + c;
    acc += expf(gate[o] - m) * emb[o];
  }
  red[rr][c] = acc; __syncthreads();
  if (rr == 0) {
    for (int i = 1; i < 8; ++i) acc += red[i][c];
    out[(size_t)g * 256 + hd * 64 + c] = acc / (s + 1e-16f);
  }
}

__global__ void edge_batch_out(const int* __restrict__ ei0, float* __restrict__ out)
{
  int e = blockIdx.x * blockDim.x + threadIdx.x;
  if (e >= EE) return;
  out[e] = (float)(ei0[e] / NPG);
}

// ===========================================================================
extern "C" void kernel_launch(void* const* d_in, const int* in_sizes, int n_in,
                              void* d_out, int out_size, void* d_ws, size_t ws_size,
                              hipStream_t stream)
{
  const float* x        = (const float*)d_in[0];
  const float* eattr    = (const float*)d_in[1];
  const float* W1       = (const float*)d_in[2];
  const float* as1      = (const float*)d_in[3];
  const float* ad1      = (const float*)d_in[4];
  const float* We1      = (const float*)d_in[5];
  const float* ae1      = (const float*)d_in[6];
  const float* bias1    = (const float*)d_in[7];
  const float* W2       = (const float*)d_in[8];
  const float* as2      = (const float*)d_in[9];
  const float* ad2      = (const float*)d_in[10];
  const float* We2      = (const float*)d_in[11];
  const float* ae2      = (const float*)d_in[12];
  const float* bias2    = (const float*)d_in[13];
  const float* Wp       = (const float*)d_in[14];
  const float* asp      = (const float*)d_in[15];
  const float* adp      = (const float*)d_in[16];
  const float* biasp    = (const float*)d_in[17];
  const float* gate_W   = (const float*)d_in[18];
  const float* gate_b   = (const float*)d_in[19];
  const float* gate_g   = (const float*)d_in[20];
  const float* gate_be  = (const float*)d_in[21];
  const float* nn_W     = (const float*)d_in[22];
  const float* nn_b     = (const float*)d_in[23];
  const float* nn_g     = (const float*)d_in[24];
  const float* nn_be    = (const float*)d_in[25];
  const int*   src0     = (const int*)d_in[26];
  const int*   dst0     = src0 + EE;

  float* out = (float*)d_out;
  float* out_gemb  = out + (size_t)EE * 4;
  float* out_batch = out_gemb + 64 * 4 * 64;

  // -------- workspace arena (floats) --------
  float* W        = (float*)d_ws;
  float* buf_h    = W;                              // N*256
  float* buf_agg  = W + (size_t)NN * 256;           // N*256
  float* buf_x2   = W + (size_t)NN * 256 * 2;       // N*256
  float* buf_x1   = W + (size_t)NN * 256 * 3;       // N*64
  float* buf_easrc= buf_x1 + (size_t)NN * 64;       // N*4
  float* buf_eadst= buf_easrc + (size_t)NN * 4;     // N*4
  unsigned* buf_mx= (unsigned*)(buf_eadst + (size_t)NN * 4);  // N*4
  float* buf_sum  = (float*)buf_mx + (size_t)NN * 4;          // N*4
  float* buf_alpha= buf_sum + (size_t)NN * 4;       // E*4
  float* buf_weff = buf_alpha + (size_t)EE * 4;     // 256
  float* buf_part = buf_weff + 256;                 // 200*128
  float* buf_bn   = buf_part + 200 * 128;           // 128
  int*   buf_ei   = (int*)(buf_bn + 128);           // 2*E ints
  // stage-C aliases (buf_h / buf_agg are dead by then)
  float* c_xh    = buf_h;
  float* c_m0    = buf_h + (size_t)NN * 64;
  float* c_m1    = buf_h + (size_t)NN * 64 * 2;
  float* c_gate  = buf_h + (size_t)NN * 64 * 3;
  float* c_emb   = buf_agg;
  float* c_hp    = buf_agg + (size_t)NN * 64;
  float* c_score = c_hp + NN;
  float* c_wself = c_score + NN;
  int*   c_perm  = (int*)(c_wself + NN);
  int*   c_inv   = c_perm + NN;

  const int TPB = 256;
  dim3 gemm_blk(128);

  // =================== GAT layer 1 ===================
  wmma_gemm_f32<<<dim3(NN/16, 4), gemm_blk, 0, stream>>>(x, W1, nullptr, buf_h, 128, 256, 128, 256);
  att_dots<<<(NN*4+TPB-1)/TPB, TPB, 0, stream>>>(buf_h, as1, ad1, buf_easrc, buf_eadst);
  weff_kernel<<<1, 256, 0, stream>>>(We1, ae1, buf_weff);
  hipMemsetAsync(buf_mx, 0, (size_t)NN*4*sizeof(unsigned), stream);
  hipMemsetAsync(buf_sum, 0, (size_t)NN*4*sizeof(float), stream);
  edge_alpha_max<<<(EE+TPB-1)/TPB, TPB, 0, stream>>>(eattr, buf_weff, src0, dst0,
                                                     buf_easrc, buf_eadst, buf_alpha, buf_mx);
  edge_expsum<<<(EE*4+TPB-1)/TPB, TPB, 0, stream>>>(buf_alpha, dst0, buf_mx, buf_sum);
  hipMemsetAsync(buf_agg, 0, (size_t)NN*256*sizeof(float), stream);
  edge_aggregate<<<(int)(((size_t)EE*64+TPB-1)/TPB), TPB, 0, stream>>>(buf_alpha, buf_sum,
                                                     src0, dst0, buf_h, buf_agg);
  x1_kernel<<<(NN*64+TPB-1)/TPB, TPB, 0, stream>>>(buf_agg, bias1, buf_x1);

  // =================== GAT layer 2 ===================
  wmma_gemm_f32<<<dim3(NN/16, 4), gemm_blk, 0, stream>>>(buf_x1, W2, nullptr, buf_agg, 64, 256, 64, 256);
  att_dots<<<(NN*4+TPB-1)/TPB, TPB, 0, stream>>>(buf_agg, as2, ad2, buf_easrc, buf_eadst);
  weff_kernel<<<1, 256, 0, stream>>>(We2, ae2, buf_weff);
  hipMemsetAsync(buf_mx, 0, (size_t)NN*4*sizeof(unsigned), stream);
  hipMemsetAsync(buf_sum, 0, (size_t)NN*4*sizeof(float), stream);
  edge_alpha_max<<<(EE+TPB-1)/TPB, TPB, 0, stream>>>(eattr, buf_weff, src0, dst0,
                                                     buf_easrc, buf_eadst, buf_alpha, buf_mx);
  edge_expsum<<<(EE*4+TPB-1)/TPB, TPB, 0, stream>>>(buf_alpha, dst0, buf_mx, buf_sum);
  hipMemsetAsync(buf_h, 0, (size_t)NN*256*sizeof(float), stream);
  edge_aggregate<<<(int)(((size_t)EE*64+TPB-1)/TPB), TPB, 0, stream>>>(buf_alpha, buf_sum,
                                                     src0, dst0, buf_agg, buf_h);
  x2_kernel<<<(int)(((size_t)NN*256+TPB-1)/TPB), TPB, 0, stream>>>(buf_h, bias2, buf_x2);

  // =================== stage C: heads ===================
  hipMemcpyAsync(buf_ei, src0, (size_t)2*EE*sizeof(int), hipMemcpyDeviceToDevice, stream);
  int* ei0 = buf_ei;
  int* ei1 = buf_ei + EE;

  auto run_mlp = [&](const float* in0, const float* Wm, const float* bm,
                     const float* gm, const float* bt, float* outbuf) {
    const float* in = in0;
    for (int l = 0; l < 4; ++l) {
      wmma_gemm_f32<<<dim3(NN/16, 1), gemm_blk, 0, stream>>>(in, Wm + l*4096, bm + l*64,
                                                             c_m1, 64, 64, 64, 64);
      col_stats_partial<<<200, 256, 0, stream>>>(c_m1, buf_part);
      bn_finalize<<<1, 64, 0, stream>>>(buf_part, buf_bn);
      bn_apply<<<(NN*64+TPB-1)/TPB, TPB, 0, stream>>>(c_m1, buf_bn, gm + l*64, bt + l*64, c_m0);
      in = c_m0;
    }
    wmma_gemm_f32<<<dim3(NN/16, 1), gemm_blk, 0, stream>>>(in, Wm + 4*4096, bm + 4*64,
                                                           outbuf, 64, 64, 64, 64);
  };

  for (int hd = 0; hd < 4; ++hd) {
    hp_kernel<<<(NN+TPB-1)/TPB, TPB, 0, stream>>>(buf_x2, Wp, hd, c_hp);
    hipMemsetAsync(buf_mx, 0, (size_t)NN*sizeof(unsigned), stream);
    hipMemsetAsync(buf_sum, 0, (size_t)NN*sizeof(float), stream);
    pool_edge_max<<<(EE+TPB-1)/TPB, TPB, 0, stream>>>(ei0, ei1, c_hp, asp, adp, buf_alpha, buf_mx);
    pool_self_max<<<(NN+TPB-1)/TPB, TPB, 0, stream>>>(c_hp, asp, adp, c_wself, buf_mx);
    pool_edge_expsum<<<(EE+TPB-1)/TPB, TPB, 0, stream>>>(buf_alpha, ei1, buf_mx, buf_sum);
    pool_self_expsum<<<(NN+TPB-1)/TPB, TPB, 0, stream>>>(c_wself, buf_mx, buf_sum);
    pool_score_init<<<(NN+TPB-1)/TPB, TPB, 0, stream>>>(c_hp, c_wself, buf_sum, biasp, c_score);
    pool_score_edges<<<(EE+TPB-1)/TPB, TPB, 0, stream>>>(buf_alpha, buf_sum, ei0, ei1, c_hp, c_score);
    edge_score_out<<<(EE+TPB-1)/TPB, TPB, 0, stream>>>(c_score, ei0, ei1, out, hd);
    seg_sort<<<NG, 1024, 0, stream>>>(c_score, c_perm, c_inv);
    xhead_kernel<<<(NN*64+TPB-1)/TPB, TPB, 0, stream>>>(buf_x2, c_score, c_perm, hd, c_xh);
    ei_update<<<(2*EE+TPB-1)/TPB, TPB, 0, stream>>>(buf_ei, c_inv);
    run_mlp(c_xh, gate_W, gate_b, gate_g, gate_be, c_gate);
    run_mlp(c_xh, nn_W,   nn_b,   nn_g,   nn_be,   c_emb);
    graph_pool<<<NG, dim3(64, 8), 0, stream>>>(c_gate, c_emb, out_gemb, hd);
  }
  edge_batch_out<<<(EE+TPB-1)/TPB, TPB, 0, stream>>>(ei0, out_batch);
}